// GHMC_Loss_32615981646270
// MI455X (gfx1250) — compile-verified
//
#include <hip/hip_runtime.h>
#include <stdint.h>

typedef __attribute__((ext_vector_type(2))) float v2f;
typedef __attribute__((ext_vector_type(8))) float v8f;

#define GHMC_BINS  10
#define THREADS    256          // 8 waves of 32
#define TILE_ELEMS 1024         // THREADS * 4 f32 per lane (one b128 per lane per array)

#if __has_builtin(__builtin_amdgcn_wmma_f32_16x16x4_f32)
#define GHMC_HAVE_WMMA_F32 1
#else
#define GHMC_HAVE_WMMA_F32 0
#endif

// ---- CDNA5 async global->LDS (ASYNCcnt path), inline asm (toolchain-portable) ----
__device__ __forceinline__ void ghmc_async_b128(uint32_t lds_off, const void* gaddr) {
  // VDST = LDS byte offset (low 32 bits of generic shared address), VADDR = 64-bit global addr
  asm volatile("global_load_async_to_lds_b128 %0, %1, off"
               :: "v"(lds_off), "v"(gaddr)
               : "memory");
}
__device__ __forceinline__ void ghmc_wait_async_le3() { asm volatile("s_wait_asynccnt 3" ::: "memory"); }
__device__ __forceinline__ void ghmc_wait_async_0()  { asm volatile("s_wait_asynccnt 0" ::: "memory"); }
__device__ __forceinline__ void ghmc_wait_ds_0()     { asm volatile("s_wait_dscnt 0"    ::: "memory"); }

__global__ void ghmc_init_ws(float* __restrict__ ws) {
  int t = threadIdx.x;
  if (t < 32) ws[t] = 0.0f;
}

// Pass 1: per-bin (valid count, bce sum) accumulated into ws[0..9] / ws[10..19].
__global__ void __launch_bounds__(THREADS, 2)
ghmc_partial_kernel(const float* __restrict__ inp,
                    const float* __restrict__ tgt,
                    const int*   __restrict__ msk,
                    float* __restrict__ ws,
                    long long n) {
  __shared__ float4 sIn[2][THREADS];   // 8 KB
  __shared__ float4 sTg[2][THREADS];   // 8 KB
  __shared__ int4   sMk[2][THREADS];   // 8 KB
  __shared__ float  P[8][20];          // per-wave partials: [wave][stat], stat = bin(cnt) or 10+bin(bce)

  const int tid  = threadIdx.x;
  const int lane = tid & 31;           // wave32
  const int wave = tid >> 5;

  float cnt[GHMC_BINS], bs[GHMC_BINS];
#pragma unroll
  for (int b = 0; b < GHMC_BINS; ++b) { cnt[b] = 0.0f; bs[b] = 0.0f; }

  const long long numTiles   = (n + TILE_ELEMS - 1) / (long long)TILE_ELEMS;
  const long long tileStride = gridDim.x;
  const long long tile0      = blockIdx.x;

  // Issue one tile's 3 async b128 loads (wave-uniform: OOB lanes clamp their address;
  // correctness is restored in compute via the per-element valid guard).
  auto issue = [&](long long t, int b) {
    long long base = t * (long long)TILE_ELEMS + (long long)tid * 4;
    if (base + 4 > n) base = n - 4;
    if (base < 0)     base = 0;
    ghmc_async_b128((uint32_t)(uintptr_t)&sIn[b][tid], (const void*)(inp + base));
    ghmc_async_b128((uint32_t)(uintptr_t)&sTg[b][tid], (const void*)(tgt + base));
    ghmc_async_b128((uint32_t)(uintptr_t)&sMk[b][tid], (const void*)(msk + base));
  };

  if (tile0 < numTiles) issue(tile0, 0);
  int buf = 0;

  for (long long t = tile0; t < numTiles; t += tileStride) {
    const long long nt = t + tileStride;
    if (nt < numTiles) {
      ghmc_wait_ds_0();          // prior reads of buf^1 must retire before DMA overwrites it
      issue(nt, buf ^ 1);        // prefetch next tile
      ghmc_wait_async_le3();     // current tile's 3 loads have landed in LDS
    } else {
      ghmc_wait_async_0();
    }

    const long long base0 = t * (long long)TILE_ELEMS + (long long)tid * 4;
    const bool chunkFull = (base0 + 4 <= n);

    float4 vx = sIn[buf][tid];
    float4 vt = sTg[buf][tid];
    int4   vm = sMk[buf][tid];
    float xs[4] = {vx.x, vx.y, vx.z, vx.w};
    float ts[4] = {vt.x, vt.y, vt.z, vt.w};
    int   ms[4] = {vm.x, vm.y, vm.z, vm.w};

#pragma unroll
    for (int j = 0; j < 4; ++j) {
      float x, tg; int m;
      if (chunkFull) {
        x = xs[j]; tg = ts[j]; m = ms[j];
      } else {
        // ragged tail (at most one lane-chunk): fall back to direct loads
        long long idx = base0 + j;
        if (idx < n) { x = inp[idx]; tg = tgt[idx]; m = msk[idx]; }
        else         { x = 0.0f;     tg = 0.0f;     m = 0;        }
      }
      float validf = (m > 0) ? 1.0f : 0.0f;

      // g = |sigmoid(x) - t| ; bin = clip(floor(g*10), 0, 9)
      float e   = __expf(-x);
      float sg  = __builtin_amdgcn_rcpf(1.0f + e);
      float g   = fabsf(sg - tg);
      int   bin = (int)(g * 10.0f);
      bin = bin > (GHMC_BINS - 1) ? (GHMC_BINS - 1) : bin;

      // stable BCE-with-logits
      float bce = fmaxf(x, 0.0f) - x * tg + __logf(1.0f + __expf(-fabsf(x)));

      // register bucketing: ~4 VALU per bin (cmp + cndmask + add + fma)
#pragma unroll
      for (int b = 0; b < GHMC_BINS; ++b) {
        float f = (bin == b) ? validf : 0.0f;
        cnt[b] += f;
        bs[b]   = fmaf(f, bce, bs[b]);
      }
    }
    buf ^= 1;
  }

  // ---- lane reduction within each wave (wave32) ----
#pragma unroll
  for (int b = 0; b < GHMC_BINS; ++b) {
#pragma unroll
    for (int off = 16; off > 0; off >>= 1) {
      cnt[b] += __shfl_down(cnt[b], off, 32);
      bs[b]  += __shfl_down(bs[b],  off, 32);
    }
  }
  if (lane == 0) {
#pragma unroll
    for (int b = 0; b < GHMC_BINS; ++b) {
      P[wave][b]             = cnt[b];
      P[wave][GHMC_BINS + b] = bs[b];
    }
  }
  __syncthreads();

  // ---- block reduction of the 8x20 partial matrix ----
  if (wave == 0) {
#if GHMC_HAVE_WMMA_F32
    // Exact f32 reduction via V_WMMA_F32_16X16X4_F32:
    //   A[m][k] = P[k][m] (stats m=0..15, waves k), B = ones(4x16) -> D[m][*] = sum_k A[m][k]
    // Two chained WMMAs cover waves 0..3 and 4..7. EXEC is all-ones here (uniform branch).
    const int m  = lane & 15;
    const int k0 = (lane < 16) ? 0 : 2;  // 32-bit A 16x4 layout: VGPR0=K0|K2, VGPR1=K1|K3
    v2f a0; a0.x = P[k0 + 0][m]; a0.y = P[k0 + 1][m];
    v2f a1; a1.x = P[k0 + 4][m]; a1.y = P[k0 + 5][m];
    v2f bo; bo.x = 1.0f;         bo.y = 1.0f;
    v8f c = {};
    c = __builtin_amdgcn_wmma_f32_16x16x4_f32(false, a0, false, bo, (short)0, c, false, false);
    c = __builtin_amdgcn_wmma_f32_16x16x4_f32(false, a1, false, bo, (short)0, c, false, false);
    // D layout: lane0 holds rows 0..7 (col 0), lane16 holds rows 8..15 (col 0)
    if (lane == 0) {
#pragma unroll
      for (int r = 0; r < 8; ++r) atomicAdd(&ws[r], c[r]);
    } else if (lane == 16) {
#pragma unroll
      for (int r = 0; r < 8; ++r) atomicAdd(&ws[8 + r], c[r]);
    } else if (lane >= 1 && lane <= 4) {
      // stats 16..19 (bce sums of bins 6..9) reduced scalarly
      int j = 15 + lane;
      float s = 0.0f;
      for (int w = 0; w < 8; ++w) s += P[w][j];
      atomicAdd(&ws[j], s);
    }
#else
    if (lane < 20) {
      float s = 0.0f;
      for (int w = 0; w < 8; ++w) s += P[w][lane];
      atomicAdd(&ws[lane], s);
    }
#endif
  }
}

// Pass 2: loss = (1/n_nonempty) * sum_b bce_sum[b] / counts[b]   (tot cancels exactly)
__global__ void ghmc_finalize_kernel(const float* __restrict__ ws, float* __restrict__ out) {
  if (blockIdx.x == 0 && threadIdx.x == 0) {
    float nbin = 0.0f, acc = 0.0f;
    for (int b = 0; b < GHMC_BINS; ++b) {
      float c = ws[b];
      if (c > 0.0f) { nbin += 1.0f; acc += ws[GHMC_BINS + b] / c; }
    }
    out[0] = (nbin > 0.0f) ? (acc / nbin) : 0.0f;
  }
}

extern "C" void kernel_launch(void* const* d_in, const int* in_sizes, int n_in,
                              void* d_out, int out_size, void* d_ws, size_t ws_size,
                              hipStream_t stream) {
  (void)n_in; (void)out_size; (void)ws_size;
  const float* inp = (const float*)d_in[0];
  const float* tgt = (const float*)d_in[1];
  const int*   msk = (const int*)d_in[2];
  float*       out = (float*)d_out;
  float*       ws  = (float*)d_ws;

  const long long n        = (long long)in_sizes[0];   // B*C = 20,971,520
  const long long numTiles = (n + TILE_ELEMS - 1) / (long long)TILE_ELEMS;
  int grid = (int)(numTiles < 1024 ? numTiles : 1024);
  if (grid < 1) grid = 1;

  ghmc_init_ws<<<1, 32, 0, stream>>>(ws);
  ghmc_partial_kernel<<<grid, THREADS, 0, stream>>>(inp, tgt, msk, ws, n);
  ghmc_finalize_kernel<<<1, 32, 0, stream>>>(ws, out);
}